// AngelLoss_83794811945269
// MI455X (gfx1250) — compile-verified
//
#include <hip/hip_runtime.h>
#include <math.h>

typedef float v2f __attribute__((ext_vector_type(2)));
typedef float v8f __attribute__((ext_vector_type(8)));

// ---------------- wave32 reduction ----------------
__device__ __forceinline__ float waveRed32(float v) {
#pragma unroll
    for (int off = 16; off > 0; off >>= 1)
        v += __shfl_xor(v, off, 32);
    return v;
}

// ---------------- 1) reciprocal center norms ----------------
// one wave per class row: rdist[c] = 1/||centers[c]||
__global__ void angel_dist_kernel(const float* __restrict__ centers,
                                  float* __restrict__ rdist, int C, int D) {
    int wave = (blockIdx.x * blockDim.x + threadIdx.x) >> 5;
    int lane = threadIdx.x & 31;
    if (wave >= C) return;  // wave-uniform exit
    const float* row = centers + (size_t)wave * D;
    float s = 0.f;
    int D4 = D >> 2;
    const float4* r4 = (const float4*)row;
    for (int i = lane; i < D4; i += 32) {
        float4 a = r4[i];
        s += a.x * a.x + a.y * a.y + a.z * a.z + a.w * a.w;
    }
    for (int d = D4 * 4 + lane; d < D; d += 32) {  // tail if D%4 != 0
        float a = row[d];
        s += a * a;
    }
    s = waveRed32(s);
    if (lane == 0) rdist[wave] = 1.0f / sqrtf(s);
}

// ---------------- 2) center loss partials ----------------
// one wave per sample row, 8 rows per 256-thread block (8 waves on CDNA5)
__global__ void angel_cen_kernel(const int* __restrict__ y,
                                 const float* __restrict__ feat,
                                 const float* __restrict__ centers,
                                 float* __restrict__ partial, int B, int D) {
    __shared__ float sred[8];
    int waveInBlk = threadIdx.x >> 5;
    int lane = threadIdx.x & 31;
    int row = blockIdx.x * 8 + waveInBlk;
    float s = 0.f;
    if (row < B) {
        int cls = y[row];
        const float* f = feat + (size_t)row * D;
        const float* c = centers + (size_t)cls * D;
        int D4 = D >> 2;
        const float4* f4 = (const float4*)f;
        const float4* c4 = (const float4*)c;
        for (int i = lane; i < D4; i += 32) {
            float4 a = f4[i];
            float4 b = c4[i];
            float dx = a.x - b.x, dy = a.y - b.y, dz = a.z - b.z, dw = a.w - b.w;
            s += dx * dx + dy * dy + dz * dz + dw * dw;
        }
        for (int d = D4 * 4 + lane; d < D; d += 32) {
            float t = f[d] - c[d];
            s += t * t;
        }
    }
    s = waveRed32(s);
    if (lane == 0) sred[waveInBlk] = s;
    __syncthreads();
    if (threadIdx.x == 0) {
        float acc = 0.f;
#pragma unroll
        for (int i = 0; i < 8; i++) acc += sred[i];
        partial[blockIdx.x] = acc;  // fixed order -> deterministic
    }
}

// ---------------- 3) angular loss via fp32 WMMA ----------------
// gram = centers @ centers^T ; one wave computes one 16x16 tile with
// v_wmma_f32_16x16x4_f32, K-loop of D/4 steps. A-frag per lane: row
// (tile_i*16 + lane%16), K pair offset (lane<16 ? 0 : 2). B = centers^T so
// B-frag is the symmetric read from the column's row. Both are float2 loads.
// Padding rows are clamped (no EXEC divergence around WMMA) and their
// contributions discarded post-accumulation.
__global__ void angel_ang_kernel(const float* __restrict__ centers,
                                 const float* __restrict__ rdist,
                                 float* __restrict__ partial, int C, int D,
                                 int T, int numTiles, float ct) {
    __shared__ float sred[8];
    int waveInBlk = threadIdx.x >> 5;
    int lane = threadIdx.x & 31;
    int tile = blockIdx.x * 8 + waveInBlk;
    float local = 0.f;
    if (tile < numTiles) {  // wave-uniform
        int ti = tile / T;
        int tj = tile % T;
        int half = lane >> 4;   // K-pair selector
        int lmod = lane & 15;   // M (for A) / N (for B)
        int arow = ti * 16 + lmod; if (arow >= C) arow = C - 1;
        int bcol = tj * 16 + lmod; if (bcol >= C) bcol = C - 1;
        const float* Ar = centers + (size_t)arow * D + (half << 1);
        const float* Br = centers + (size_t)bcol * D + (half << 1);
        v8f acc = {0.f, 0.f, 0.f, 0.f, 0.f, 0.f, 0.f, 0.f};
#pragma unroll 8
        for (int k = 0; k < D; k += 4) {
            v2f av = *(const v2f*)(Ar + k);
            v2f bv = *(const v2f*)(Br + k);
            // 8 args: (neg_a, A, neg_b, B, c_mod, C, reuse_a, reuse_b)
            acc = __builtin_amdgcn_wmma_f32_16x16x4_f32(
                false, av, false, bv, (short)0, acc, false, false);
        }
        // C/D layout: lanes 0-15 -> M = v, lanes 16-31 -> M = v + 8; N = lane%16
        int j = tj * 16 + lmod;
        float rdj = rdist[bcol];  // == rdist[j] whenever j < C
#pragma unroll
        for (int v = 0; v < 8; v++) {
            int i = ti * 16 + v + (half << 3);
            if (i < C && j < C && i != j) {
                float sim = acc[v] * rdist[i] * rdj;
                float t = sim - ct;
                local += t * t;
            }
        }
    }
    local = waveRed32(local);
    if (lane == 0) sred[waveInBlk] = local;
    __syncthreads();
    if (threadIdx.x == 0) {
        float acc = 0.f;
#pragma unroll
        for (int i = 0; i < 8; i++) acc += sred[i];
        partial[blockIdx.x] = acc;
    }
}

// ---------------- 4) deterministic final reduction ----------------
__global__ void angel_final_kernel(const float* __restrict__ cenP, int nCen,
                                   const float* __restrict__ angP, int nAng,
                                   float* __restrict__ out, float cenScale,
                                   float angScale) {
    __shared__ float sred[256];
    float s = 0.f;
    for (int i = threadIdx.x; i < nCen; i += 256) s += cenP[i] * cenScale;
    for (int i = threadIdx.x; i < nAng; i += 256) s += angP[i] * angScale;
    sred[threadIdx.x] = s;
    __syncthreads();
#pragma unroll
    for (int off = 128; off > 0; off >>= 1) {
        if (threadIdx.x < off) sred[threadIdx.x] += sred[threadIdx.x + off];
        __syncthreads();
    }
    if (threadIdx.x == 0) out[0] = sred[0];
}

extern "C" void kernel_launch(void* const* d_in, const int* in_sizes, int n_in,
                              void* d_out, int out_size, void* d_ws,
                              size_t ws_size, hipStream_t stream) {
    const int* y = (const int*)d_in[0];
    const float* feat = (const float*)d_in[1];
    const float* centers = (const float*)d_in[2];

    const int B = in_sizes[0];
    const int D = in_sizes[1] / B;
    const int C = in_sizes[2] / D;

    // static ct = 2*radius(C-1)^2 - 1, radius via iterative recursion (host, fp64)
    double rad;
    {
        int nd = C - 1;
        if (nd <= 1) {
            rad = 0.0;
        } else {
            rad = 0.5;
            for (int i = 3; i <= nd; i++) rad = 0.5 / sqrt(1.0 - rad * rad);
        }
    }
    const float ct = (float)(2.0 * rad * rad - 1.0);

    // workspace layout (floats): [rdist C][cen partials][ang partials]
    float* ws = (float*)d_ws;
    float* rdist = ws;
    const int nCen = (B + 7) / 8;
    float* cenP = rdist + C;
    const int T = (C + 15) / 16;
    const int numTiles = T * T;
    const int nAng = (numTiles + 7) / 8;
    float* angP = cenP + nCen;

    angel_dist_kernel<<<(C + 7) / 8, 256, 0, stream>>>(centers, rdist, C, D);
    angel_cen_kernel<<<nCen, 256, 0, stream>>>(y, feat, centers, cenP, B, D);
    angel_ang_kernel<<<nAng, 256, 0, stream>>>(centers, rdist, angP, C, D, T,
                                               numTiles, ct);
    const float cenScale = 0.5f / (float)B;                      // * CEN_W (1.0)
    const float angScale = 1.0f / (0.5f * (float)C * ((float)C - 1.0f));  // * ANG_W
    angel_final_kernel<<<1, 256, 0, stream>>>(cenP, nCen, angP, nAng,
                                              (float*)d_out, cenScale, angScale);
}